// QuantumHybrid_65481071399358
// MI455X (gfx1250) — compile-verified
//
#include <hip/hip_runtime.h>
#include <hip/hip_bf16.h>
#include <math.h>

// ---------------------------------------------------------------------------
// 4-qubit hybrid quantum circuit, MI455X (gfx1250, wave32, WMMA).
//  K1: build the fixed post-encoder 16x16 complex unitary in LDS.
//  K2: pool + encode (product state) + U_post matvec via V_WMMA_F32_16X16X4_F32
//      + Pauli-Z expectations + batch-stat partial sums.
//  K3: batchnorm + fc.
// ---------------------------------------------------------------------------

typedef __attribute__((ext_vector_type(2))) float v2f;
typedef __attribute__((ext_vector_type(8))) float v8f;

__device__ __forceinline__ float2 cmul(float2 a, float2 b) {
    return make_float2(a.x * b.x - a.y * b.y, a.x * b.y + a.y * b.x);
}
__device__ __forceinline__ float2 cadd(float2 a, float2 b) {
    return make_float2(a.x + b.x, a.y + b.y);
}

// ---- gate program (post-encoder): 50 "random layer" ops + trainable tail ----
// kinds: 0=RX 1=RY 2=RZ 3=CNOT 4=CRX 5=H 6=SX
#define N_OPS 57
__device__ __constant__ signed char OPK[N_OPS] = {
    0,1,3,2,1,0,3,2,1,3, 0,2,1,3,0,1,2,3,0,1,
    3,2,0,1,3,2,0,1,3,2, 0,1,3,2,0,1,3,2,0,1,
    3,2,0,1,3,2,0,1,3,2,
    0,1,2,4,5,6,3};
__device__ __constant__ signed char OPW0[N_OPS] = {
    0,2,1,3,0,2,0,1,3,2, 1,0,2,3,3,1,2,1,0,3,
    2,0,2,1,0,3,1,0,3,2, 3,2,1,1,0,3,0,0,2,1,
    2,3,1,2,3,2,3,0,1,1,
    0,1,3,0,3,2,3};
__device__ __constant__ signed char OPW1[N_OPS] = {
    0,0,3,0,0,0,2,0,0,0, 0,0,0,1,0,0,0,0,0,0,
    3,0,0,0,1,0,0,0,2,0, 0,0,2,0,0,0,3,0,0,0,
    1,0,0,0,0,0,0,0,3,0,
    0,0,0,2,0,0,0};
__device__ __constant__ short OPP[N_OPS] = {
    0,1,-1,2,3,4,-1,5,6,-1, 7,8,9,-1,10,11,12,-1,13,14,
    -1,15,16,17,-1,18,19,20,-1,21, 22,23,-1,24,25,26,-1,27,28,29,
    -1,30,31,32,-1,33,34,35,-1,36,
    1000,1001,1002,1003,-1,-1,-1};

// ---------------------------------------------------------------------------
// K1: build U_post (16x16 complex) by left-multiplying gates.  1 block x 128.
// ws layout (floats): [0..255]=Ur  [256..511]=Ui  [512..767]=-Ui  [768..775]=sums
// ---------------------------------------------------------------------------
__global__ void __launch_bounds__(128)
build_unitary(const float* __restrict__ qparams,
              const float* __restrict__ rparams, int nrand,
              float* __restrict__ ws) {
    __shared__ float2 U[16][16];
    const int tid = threadIdx.x;

    // U = I
    for (int e = tid; e < 256; e += 128) {
        const int i = e >> 4, j = e & 15;
        U[i][j] = make_float2(i == j ? 1.f : 0.f, 0.f);
    }
    __syncthreads();

    const int p = tid >> 4;   // row-pair 0..7
    const int j = tid & 15;   // column

    for (int op = 0; op < N_OPS; ++op) {
        const int kind = OPK[op];
        int tgt = OPW0[op], ctrl = -1;

        float t = 0.f;
        const int pi = OPP[op];
        if (pi >= 1000) t = qparams[pi - 1000];
        else if (pi >= 0) {
            int idx = pi < nrand ? pi : (nrand > 0 ? nrand - 1 : 0);
            t = rparams[idx];
        }
        const float ch = cosf(0.5f * t), sh = sinf(0.5f * t);

        float2 g00, g01, g10, g11;
        switch (kind) {
        case 0:  // RX
            g00 = {ch, 0.f}; g01 = {0.f, -sh}; g10 = {0.f, -sh}; g11 = {ch, 0.f};
            break;
        case 1:  // RY
            g00 = {ch, 0.f}; g01 = {-sh, 0.f}; g10 = {sh, 0.f}; g11 = {ch, 0.f};
            break;
        case 2:  // RZ
            g00 = {ch, -sh}; g01 = {0.f, 0.f}; g10 = {0.f, 0.f}; g11 = {ch, sh};
            break;
        case 3:  // CNOT = controlled-X
            g00 = {0.f, 0.f}; g01 = {1.f, 0.f}; g10 = {1.f, 0.f}; g11 = {0.f, 0.f};
            ctrl = OPW0[op]; tgt = OPW1[op];
            break;
        case 4:  // CRX
            g00 = {ch, 0.f}; g01 = {0.f, -sh}; g10 = {0.f, -sh}; g11 = {ch, 0.f};
            ctrl = OPW0[op]; tgt = OPW1[op];
            break;
        case 5: { // H
            const float s = 0.70710678118654752f;
            g00 = {s, 0.f}; g01 = {s, 0.f}; g10 = {s, 0.f}; g11 = {-s, 0.f};
        } break;
        default: // SX
            g00 = {0.5f, 0.5f}; g01 = {0.5f, -0.5f};
            g10 = {0.5f, -0.5f}; g11 = {0.5f, 0.5f};
            break;
        }

        const int tb = 3 - tgt;                       // wire 0 = MSB
        const int cb = (ctrl >= 0) ? (3 - ctrl) : -1;
        const int i0 = ((p >> tb) << (tb + 1)) | (p & ((1 << tb) - 1));
        const int i1 = i0 | (1 << tb);

        const float2 u0 = U[i0][j], u1 = U[i1][j];
        __syncthreads();
        const bool act = (cb < 0) || (((i0 >> cb) & 1) != 0);
        const float2 n0 = act ? cadd(cmul(g00, u0), cmul(g01, u1)) : u0;
        const float2 n1 = act ? cadd(cmul(g10, u0), cmul(g11, u1)) : u1;
        U[i0][j] = n0;
        U[i1][j] = n1;
        __syncthreads();
    }

    for (int e = tid; e < 256; e += 128) {
        const int i = e >> 4, jj = e & 15;
        const float2 u = U[i][jj];
        ws[e]       = u.x;   // Ur
        ws[256 + e] = u.y;   // Ui
        ws[512 + e] = -u.y;  // -Ui
    }
    if (tid < 8) ws[768 + tid] = 0.f;  // batchnorm accumulators
}

// ---------------------------------------------------------------------------
// K2: per-sample pool + encode + WMMA matvec + <Z> + batch partial sums.
// 256 threads (8 waves) = 256 samples per block; 16 samples per WMMA tile.
// ---------------------------------------------------------------------------
__global__ void __launch_bounds__(256)
qsim_main(const float* __restrict__ inputs,
          const float* __restrict__ ws,     // Ur/Ui/-Ui + sums
          float* __restrict__ exps,         // [4][nsamp]
          int nsamp) {
    __shared__ float sRe[256][17];
    __shared__ float sIm[256][17];
    __shared__ float blkAcc[8];

    const int tid = threadIdx.x;
    const int sample = blockIdx.x * 256 + tid;
    if (tid < 8) blkAcc[tid] = 0.f;

    // ---- phase 1: avg-pool 24x24 -> 16 (b128 loads, image read exactly once)
    const float* img = inputs + (size_t)sample * 576;
    float pac[16];
#pragma unroll
    for (int i = 0; i < 16; ++i) pac[i] = 0.f;
#pragma unroll
    for (int row = 0; row < 24; ++row) {
        const int g = (row / 6) * 4;
        const float4* rp = reinterpret_cast<const float4*>(img + row * 24);
        const float4 v0 = rp[0], v1 = rp[1], v2 = rp[2];
        const float4 v3 = rp[3], v4 = rp[4], v5 = rp[5];
        pac[g + 0] += v0.x + v0.y + v0.z + v0.w + v1.x + v1.y;
        pac[g + 1] += v1.z + v1.w + v2.x + v2.y + v2.z + v2.w;
        pac[g + 2] += v3.x + v3.y + v3.z + v3.w + v4.x + v4.y;
        pac[g + 3] += v4.z + v4.w + v5.x + v5.y + v5.z + v5.w;
    }
    const float inv36 = 1.f / 36.f;

    // ---- encoder: per-wire v = RY(d) RX(c) RZ(b) RY(a) |0>  (product state)
    float2 st0[4], st1[4];
#pragma unroll
    for (int w = 0; w < 4; ++w) {
        const float ha = 0.5f * pac[w] * inv36;
        const float hb = 0.5f * pac[4 + w] * inv36;
        const float hc = 0.5f * pac[8 + w] * inv36;
        const float hd = 0.5f * pac[12 + w] * inv36;
        const float ca = cosf(ha), sa = sinf(ha);
        const float cb = cosf(hb), sb = sinf(hb);
        const float cc = cosf(hc), sc = sinf(hc);
        const float cd = cosf(hd), sd = sinf(hd);
        const float2 v0 = {ca * cb, -ca * sb};
        const float2 v1 = {sa * cb,  sa * sb};
        const float2 u0 = {cc * v0.x + sc * v1.y,  cc * v0.y - sc * v1.x};
        const float2 u1 = {sc * v0.y + cc * v1.x, -sc * v0.x + cc * v1.y};
        st0[w] = {cd * u0.x - sd * u1.x, cd * u0.y - sd * u1.y};
        st1[w] = {sd * u0.x + cd * u1.x, sd * u0.y + cd * u1.y};
    }
    // kron -> 16 complex amplitudes into LDS (B-matrix staging)
#pragma unroll
    for (int b0 = 0; b0 < 2; ++b0) {
        const float2 a0 = b0 ? st1[0] : st0[0];
#pragma unroll
        for (int b1 = 0; b1 < 2; ++b1) {
            const float2 a01 = cmul(a0, b1 ? st1[1] : st0[1]);
#pragma unroll
            for (int b2 = 0; b2 < 2; ++b2) {
                const float2 a012 = cmul(a01, b2 ? st1[2] : st0[2]);
#pragma unroll
                for (int b3 = 0; b3 < 2; ++b3) {
                    const float2 amp = cmul(a012, b3 ? st1[3] : st0[3]);
                    const int m = (b0 << 3) | (b1 << 2) | (b2 << 1) | b3;
                    sRe[tid][m] = amp.x;
                    sIm[tid][m] = amp.y;
                }
            }
        }
    }
    __syncthreads();

    // ---- phase 2: Y = U_post * Psi via V_WMMA_F32_16X16X4_F32 (Re/Im split)
    const int lane = tid & 31;
    const int wave = tid >> 5;
    const int M = lane & 15;
    const int kb = (lane < 16) ? 0 : 2;  // ISA 7.12.2 A/B half-wave K split

    const float* Ur = ws;
    const float* Ui = ws + 256;
    const float* Un = ws + 512;

    v2f aR[4], aI[4], aN[4];
#pragma unroll
    for (int kk = 0; kk < 4; ++kk) {
        const int k = 4 * kk + kb;
        aR[kk] = v2f{Ur[M * 16 + k], Ur[M * 16 + k + 1]};
        aI[kk] = v2f{Ui[M * 16 + k], Ui[M * 16 + k + 1]};
        aN[kk] = v2f{Un[M * 16 + k], Un[M * 16 + k + 1]};
    }

#pragma unroll
    for (int t = 0; t < 2; ++t) {            // two 16-sample tiles per wave
        const int s = wave * 32 + t * 16 + (lane & 15);
        v8f yr = {0.f, 0.f, 0.f, 0.f, 0.f, 0.f, 0.f, 0.f};
        v8f yi = {0.f, 0.f, 0.f, 0.f, 0.f, 0.f, 0.f, 0.f};
#pragma unroll
        for (int kk = 0; kk < 4; ++kk) {
            const int k = 4 * kk + kb;
            const v2f br = v2f{sRe[s][k], sRe[s][k + 1]};
            const v2f bi = v2f{sIm[s][k], sIm[s][k + 1]};
            // Re: Ur*Pr + (-Ui)*Pi ; Im: Ur*Pi + Ui*Pr
            yr = __builtin_amdgcn_wmma_f32_16x16x4_f32(false, aR[kk], false, br,
                                                       (short)0, yr, false, false);
            yr = __builtin_amdgcn_wmma_f32_16x16x4_f32(false, aN[kk], false, bi,
                                                       (short)0, yr, false, false);
            yi = __builtin_amdgcn_wmma_f32_16x16x4_f32(false, aR[kk], false, bi,
                                                       (short)0, yi, false, false);
            yi = __builtin_amdgcn_wmma_f32_16x16x4_f32(false, aI[kk], false, br,
                                                       (short)0, yi, false, false);
        }
        // ---- <Z_w>: lane holds D rows M=r(+8); combine halves via shfl_xor 16
        float e0 = 0.f, e1 = 0.f, e2 = 0.f, e3 = 0.f;
        const int moff = (lane < 16) ? 0 : 8;
#pragma unroll
        for (int r = 0; r < 8; ++r) {
            const int m = r + moff;
            const float pr = yr[r], pim = yi[r];
            const float pp = pr * pr + pim * pim;
            e0 += ((m >> 3) & 1) ? -pp : pp;
            e1 += ((m >> 2) & 1) ? -pp : pp;
            e2 += ((m >> 1) & 1) ? -pp : pp;
            e3 += (m & 1) ? -pp : pp;
        }
        e0 += __shfl_xor(e0, 16, 32);
        e1 += __shfl_xor(e1, 16, 32);
        e2 += __shfl_xor(e2, 16, 32);
        e3 += __shfl_xor(e3, 16, 32);

        if (lane < 16) {
            const int gs = blockIdx.x * 256 + s;
            exps[0 * nsamp + gs] = e0;
            exps[1 * nsamp + gs] = e1;
            exps[2 * nsamp + gs] = e2;
            exps[3 * nsamp + gs] = e3;
            atomicAdd(&blkAcc[0], e0);
            atomicAdd(&blkAcc[1], e1);
            atomicAdd(&blkAcc[2], e2);
            atomicAdd(&blkAcc[3], e3);
            atomicAdd(&blkAcc[4], e0 * e0);
            atomicAdd(&blkAcc[5], e1 * e1);
            atomicAdd(&blkAcc[6], e2 * e2);
            atomicAdd(&blkAcc[7], e3 * e3);
        }
    }
    __syncthreads();
    if (tid < 8) atomicAdd(const_cast<float*>(ws) + 768 + tid, blkAcc[tid]);
}

// ---------------------------------------------------------------------------
// K3: batchnorm (batch stats, biased var, eps=1e-5) + fc  -> d_out [nsamp]
// ---------------------------------------------------------------------------
__global__ void __launch_bounds__(256)
finalize(const float* __restrict__ exps, const float* __restrict__ sums,
         const float* __restrict__ gamma, const float* __restrict__ beta,
         const float* __restrict__ fcw, const float* __restrict__ fcb,
         float* __restrict__ out, int nsamp) {
    const int s = blockIdx.x * blockDim.x + threadIdx.x;
    if (s >= nsamp) return;
    const float invn = 1.f / (float)nsamp;
    float acc = fcb[0];
#pragma unroll
    for (int w = 0; w < 4; ++w) {
        const float mean = sums[w] * invn;
        const float var = sums[4 + w] * invn - mean * mean;
        const float x = exps[w * nsamp + s];
        acc += ((x - mean) * rsqrtf(var + 1e-5f) * gamma[w] + beta[w]) * fcw[w];
    }
    out[s] = acc;
}

// ---------------------------------------------------------------------------
extern "C" void kernel_launch(void* const* d_in, const int* in_sizes, int n_in,
                              void* d_out, int out_size, void* d_ws, size_t ws_size,
                              hipStream_t stream) {
    const float* inputs  = (const float*)d_in[0];
    const float* qparams = (const float*)d_in[1];
    const float* rparams = (const float*)d_in[2];
    const float* gamma   = (const float*)d_in[3];
    const float* beta    = (const float*)d_in[4];
    const float* fcw     = (const float*)d_in[5];
    const float* fcb     = (const float*)d_in[6];

    const int nsamp = in_sizes[0] / 576;   // 65536
    const int nrand = in_sizes[2];

    float* ws   = (float*)d_ws;
    float* sums = ws + 768;
    float* exps = ws + 1024;               // [4][nsamp] floats

    build_unitary<<<1, 128, 0, stream>>>(qparams, rparams, nrand, ws);
    qsim_main<<<nsamp / 256, 256, 0, stream>>>(inputs, ws, exps, nsamp);
    finalize<<<(nsamp + 255) / 256, 256, 0, stream>>>(exps, sums, gamma, beta,
                                                      fcw, fcb, (float*)d_out,
                                                      nsamp);
}